// SigAutoEncoder_28295244546653
// MI455X (gfx1250) — compile-verified
//
#include <hip/hip_runtime.h>
#include <math.h>

typedef float v2f __attribute__((ext_vector_type(2)));
typedef float v8f __attribute__((ext_vector_type(8)));

#define WMMA_F32x4(a, b, c) \
  __builtin_amdgcn_wmma_f32_16x16x4_f32(false, (a), false, (b), (short)0, (c), false, false)

__device__ __forceinline__ float sigm(float x) { return 1.0f / (1.0f + __expf(-x)); }

__device__ __forceinline__ unsigned int pcg_hash(unsigned int v) {
  unsigned int s = v * 747796405u + 2891336453u;
  unsigned int w = ((s >> ((s >> 28u) + 4u)) ^ s) * 277803737u;
  return (w >> 22u) ^ w;
}

// ---------------------------------------------------------------------------
// Encoder conv stack (tiny): conv1+relu -> maxpool4 -> conv2+relu
// ---------------------------------------------------------------------------
__global__ void k_conv1(const float* __restrict__ x, const float* __restrict__ W,
                        const float* __restrict__ bias, float* __restrict__ y) {
  int idx = blockIdx.x * blockDim.x + threadIdx.x;  // 16*64*93
  if (idx >= 16 * 64 * 93) return;
  int p = idx % 93;
  int c = (idx / 93) % 64;
  int b = idx / (93 * 64);
  float acc = bias[c];
  const float* xb = x + b * 96 * 96;
  const float* wc = W + c * 96 * 4;
  for (int d = 0; d < 96; ++d) {
    const float* xr = xb + d * 96 + p;
    const float* wr = wc + d * 4;
    acc += xr[0] * wr[0] + xr[1] * wr[1] + xr[2] * wr[2] + xr[3] * wr[3];
  }
  y[idx] = fmaxf(acc, 0.0f);
}

__global__ void k_pool(const float* __restrict__ y, float* __restrict__ p) {
  int idx = blockIdx.x * blockDim.x + threadIdx.x;  // 16*64*23
  if (idx >= 16 * 64 * 23) return;
  int q = idx % 23;
  int bc = idx / 23;  // b*64+c
  const float* src = y + bc * 93 + q * 4;
  float m = fmaxf(fmaxf(src[0], src[1]), fmaxf(src[2], src[3]));
  p[idx] = m;
}

__global__ void k_conv2(const float* __restrict__ pool, const float* __restrict__ W,
                        const float* __restrict__ bias, float* __restrict__ henc) {
  int idx = blockIdx.x * blockDim.x + threadIdx.x;  // 16*64*20
  if (idx >= 16 * 64 * 20) return;
  int p = idx % 20;
  int c = (idx / 20) % 64;
  int b = idx / (20 * 64);
  float acc = bias[c];
  for (int cin = 0; cin < 64; ++cin) {
    const float* pr = pool + (b * 64 + cin) * 23 + p;
    const float* wr = W + (c * 64 + cin) * 4;
    acc += pr[0] * wr[0] + pr[1] * wr[1] + pr[2] * wr[2] + pr[3] * wr[3];
  }
  henc[idx] = fmaxf(acc, 0.0f);  // henc[b][t][d] : (16,64,20)
}

// ---------------------------------------------------------------------------
// Encoder GRU input projections: xwe[dir][t][b][g] (4,64,16,288), K=20 scalar
// ---------------------------------------------------------------------------
__global__ void k_xw_enc(const float* __restrict__ henc,
                         const float* Wi0, const float* Wi1, const float* Wi2, const float* Wi3,
                         const float* bi0, const float* bi1, const float* bi2, const float* bi3,
                         float* __restrict__ xwe) {
  int idx = blockIdx.x * blockDim.x + threadIdx.x;
  if (idx >= 4 * 64 * 16 * 288) return;
  int g = idx % 288;
  int b = (idx / 288) % 16;
  int t = (idx / (288 * 16)) % 64;
  int dir = idx / (288 * 16 * 64);
  const float* Wi = dir == 0 ? Wi0 : dir == 1 ? Wi1 : dir == 2 ? Wi2 : Wi3;
  const float* bi = dir == 0 ? bi0 : dir == 1 ? bi1 : dir == 2 ? bi2 : bi3;
  const float* x = henc + b * 1280 + t * 20;
  const float* w = Wi + g * 20;
  float a = bi[g];
#pragma unroll
  for (int d = 0; d < 20; ++d) a += x[d] * w[d];
  xwe[idx] = a;
}

// ---------------------------------------------------------------------------
// Encoder GRU recurrence: 4 independent chains (blocks), 64 steps each.
// H=96 -> gates 288 = 18 N-tiles; 6 waves x 3 tiles (r/z/n of same columns).
// Wh fragments resident in VGPRs; gate fusion entirely in registers.
// ---------------------------------------------------------------------------
__global__ __launch_bounds__(192, 1) void k_gru_enc(
    const float* Wh0, const float* Wh1, const float* Wh2, const float* Wh3,
    const float* bh0, const float* bh1, const float* bh2, const float* bh3,
    const float* __restrict__ xwe, float* __restrict__ outp) {
  __shared__ float hbuf[16 * 132];  // stride 132 (mod 64 == 4: conflict-free b64)
  const int dir = blockIdx.x;
  const float* Wh = dir == 0 ? Wh0 : dir == 1 ? Wh1 : dir == 2 ? Wh2 : Wh3;
  const float* bh = dir == 0 ? bh0 : dir == 1 ? bh1 : dir == 2 ? bh2 : bh3;
  const float* xwb = xwe + dir * (64 * 16 * 288);
  const int tid = threadIdx.x, lane = tid & 31, wave = tid >> 5;
  const int mrow = lane & 15;            // A-row / C-col within tile
  const int khalf = (lane >> 4) << 1;    // K offset for lane half
  const int rbase = (lane >> 4) << 3;    // C/D: M offset for lane half
  const int jj = wave * 16 + mrow;       // hidden column this lane owns

  // B fragments for 3 gate tiles (cols jj, 96+jj, 192+jj), K=96 -> 24 frags each
  v2f bf0[24], bf1[24], bf2[24];
#pragma unroll
  for (int kc = 0; kc < 24; ++kc) {
    int kk = 4 * kc + khalf;
    bf0[kc].x = Wh[(jj) * 96 + kk];        bf0[kc].y = Wh[(jj) * 96 + kk + 1];
    bf1[kc].x = Wh[(96 + jj) * 96 + kk];   bf1[kc].y = Wh[(96 + jj) * 96 + kk + 1];
    bf2[kc].x = Wh[(192 + jj) * 96 + kk];  bf2[kc].y = Wh[(192 + jj) * 96 + kk + 1];
  }
  const float bhr = bh[jj], bhz = bh[96 + jj], bhn = bh[192 + jj];

  for (int e = tid; e < 16 * 132; e += 192) hbuf[e] = 0.0f;
  __syncthreads();

  const bool rev = (dir & 1);
  for (int s = 0; s < 64; ++s) {
    v8f a0 = {0, 0, 0, 0, 0, 0, 0, 0}, a1 = a0, a2 = a0;
#pragma unroll
    for (int kc = 0; kc < 24; ++kc) {
      int kk = 4 * kc + khalf;
      v2f a;
      a.x = hbuf[mrow * 132 + kk];
      a.y = hbuf[mrow * 132 + kk + 1];
      a0 = WMMA_F32x4(a, bf0[kc], a0);
      a1 = WMMA_F32x4(a, bf1[kc], a1);
      a2 = WMMA_F32x4(a, bf2[kc], a2);
    }
    __syncthreads();  // all matmul reads of hbuf done before updates
    const int t = rev ? (63 - s) : s;
    const float* xwt = xwb + t * (16 * 288);
#pragma unroll
    for (int r = 0; r < 8; ++r) {
      int m = rbase + r;
      const float* xw = xwt + m * 288;
      float rr = sigm(xw[jj] + a0[r] + bhr);
      float zz = sigm(xw[96 + jj] + a1[r] + bhz);
      float nn = tanhf(xw[192 + jj] + rr * (a2[r] + bhn));
      float ho = hbuf[m * 132 + jj];
      hbuf[m * 132 + jj] = (1.0f - zz) * nn + zz * ho;
    }
    __syncthreads();
  }
  // final hidden -> mu / logvar slice of d_out
  float* dst = outp + 147456 + (dir >= 2 ? 3072 : 0);
  const int cb = (dir & 1) ? 96 : 0;
  for (int e = tid; e < 16 * 96; e += 192) {
    int m = e / 96, j = e % 96;
    dst[m * 192 + cb + j] = hbuf[m * 132 + j];
  }
}

// ---------------------------------------------------------------------------
// z = mu + eps * exp(0.5*logvar)  (deterministic hash-based normal eps)
// ---------------------------------------------------------------------------
__global__ void k_z(const float* __restrict__ outbuf, float* __restrict__ zbuf) {
  int e = blockIdx.x * blockDim.x + threadIdx.x;
  if (e >= 3072) return;
  float mu = outbuf[147456 + e];
  float lv = outbuf[147456 + 3072 + e];
  unsigned int r1 = pcg_hash(2u * (unsigned)e + 1u);
  unsigned int r2 = pcg_hash(2u * (unsigned)e + 2u);
  float u1 = ((float)r1 + 1.0f) * (1.0f / 4294967296.0f);
  float u2 = (float)r2 * (1.0f / 4294967296.0f);
  float eps = sqrtf(-2.0f * __logf(u1)) * __cosf(6.28318530718f * u2);
  zbuf[e] = mu + eps * __expf(0.5f * lv);
}

// ---------------------------------------------------------------------------
// Decoder input projections via WMMA: XW[(b*96+t)*576+g] = input@Wi_d^T + bi_d
// M=1536, K=96, N=576 -> 96x36 waves, each 16x16 tile with 24 K-steps.
// ---------------------------------------------------------------------------
__global__ __launch_bounds__(128) void k_xw_dec(const float* __restrict__ X,
                                                const float* __restrict__ Wid,
                                                const float* __restrict__ bid,
                                                float* __restrict__ XW) {
  const int lane = threadIdx.x & 31;
  const int wg = blockIdx.x * 4 + (threadIdx.x >> 5);  // 0..3455
  const int mt = wg / 36, nt = wg % 36;
  const int row0 = mt * 16, n0 = nt * 16;
  const int mrow = lane & 15, khalf = (lane >> 4) << 1, rbase = (lane >> 4) << 3;
  v8f acc = {0, 0, 0, 0, 0, 0, 0, 0};
#pragma unroll
  for (int kc = 0; kc < 24; ++kc) {
    int kk = 4 * kc + khalf;
    v2f a, b;
    a.x = X[(row0 + mrow) * 96 + kk];
    a.y = X[(row0 + mrow) * 96 + kk + 1];
    b.x = Wid[(n0 + mrow) * 96 + kk];
    b.y = Wid[(n0 + mrow) * 96 + kk + 1];
    acc = WMMA_F32x4(a, b, acc);
  }
  const float bc = bid[n0 + mrow];
#pragma unroll
  for (int r = 0; r < 8; ++r) XW[(row0 + rbase + r) * 576 + n0 + mrow] = acc[r] + bc;
}

// ---------------------------------------------------------------------------
// Decoder: persistent single workgroup, 9120 sequential GRU steps (H=192).
// 12 waves; wave w owns hidden cols [16w,16w+16): tiles w (r), w+12 (z),
// w+24 (n). Wh_d fully VGPR-resident (3x48 v2f frags); gates fused in regs.
// ---------------------------------------------------------------------------
__global__ __launch_bounds__(384, 1) void k_gru_dec(
    const float* __restrict__ Whd, const float* __restrict__ bhd,
    const float* __restrict__ XW, const float* __restrict__ zbuf,
    const float* __restrict__ Wl, const float* __restrict__ bl,
    float* __restrict__ out) {
  __shared__ float hbuf[16 * 196];  // stride 196 (mod 64 == 4: conflict-free b64)
  const int tid = threadIdx.x, lane = tid & 31, wave = tid >> 5;
  const int mrow = lane & 15, khalf = (lane >> 4) << 1, rbase = (lane >> 4) << 3;
  const int jj = wave * 16 + mrow;  // hidden column owned by this lane

  v2f bf0[48], bf1[48], bf2[48];
#pragma unroll
  for (int kc = 0; kc < 48; ++kc) {
    int kk = 4 * kc + khalf;
    bf0[kc].x = Whd[(jj) * 192 + kk];        bf0[kc].y = Whd[(jj) * 192 + kk + 1];
    bf1[kc].x = Whd[(192 + jj) * 192 + kk];  bf1[kc].y = Whd[(192 + jj) * 192 + kk + 1];
    bf2[kc].x = Whd[(384 + jj) * 192 + kk];  bf2[kc].y = Whd[(384 + jj) * 192 + kk + 1];
  }
  const float bhr = bhd[jj], bhz = bhd[192 + jj], bhn = bhd[384 + jj];

  for (int e = tid; e < 16 * 192; e += 384) {
    int m = e / 192, j = e % 192;
    hbuf[m * 196 + j] = zbuf[e];  // h0 = z
  }
  __syncthreads();

  for (int i = 0; i < 96; ++i) {
    for (int s = 0; s < 95; ++s) {
      const int t = s + (s >= i ? 1 : 0);  // leave-one-out timestep index
      v8f a0 = {0, 0, 0, 0, 0, 0, 0, 0}, a1 = a0, a2 = a0;
#pragma unroll
      for (int kc = 0; kc < 48; ++kc) {
        int kk = 4 * kc + khalf;
        v2f a;
        a.x = hbuf[mrow * 196 + kk];
        a.y = hbuf[mrow * 196 + kk + 1];
        a0 = WMMA_F32x4(a, bf0[kc], a0);
        a1 = WMMA_F32x4(a, bf1[kc], a1);
        a2 = WMMA_F32x4(a, bf2[kc], a2);
      }
      __syncthreads();  // matmul reads of h complete before h update
#pragma unroll
      for (int r = 0; r < 8; ++r) {
        int m = rbase + r;
        const float* xw = XW + (m * 96 + t) * 576;
        float rr = sigm(xw[jj] + a0[r] + bhr);
        float zz = sigm(xw[192 + jj] + a1[r] + bhz);
        float nn = tanhf(xw[384 + jj] + rr * (a2[r] + bhn));
        float ho = hbuf[m * 196 + jj];
        hbuf[m * 196 + jj] = (1.0f - zz) * nn + zz * ho;
      }
      __syncthreads();
    }
    // emit out[b][i][:] = h @ Wl^T + bl   (scalar dots; 56 MFLOP total)
    for (int e = tid; e < 16 * 96; e += 384) {
      int m = e / 96, o = e % 96;
      float a = bl[o];
      const float* wr = Wl + o * 192;
      for (int k = 0; k < 192; ++k) a += hbuf[m * 196 + k] * wr[k];
      out[m * 9216 + i * 96 + o] = a;
    }
  }
}

// ---------------------------------------------------------------------------
extern "C" void kernel_launch(void* const* d_in, const int* in_sizes, int n_in,
                              void* d_out, int out_size, void* d_ws, size_t ws_size,
                              hipStream_t stream) {
  (void)in_sizes; (void)n_in; (void)out_size; (void)ws_size;
  const float* input = (const float*)d_in[0];
  const float* W_c1 = (const float*)d_in[1];
  const float* b_c1 = (const float*)d_in[2];
  const float* W_c2 = (const float*)d_in[3];
  const float* b_c2 = (const float*)d_in[4];
  const float* Wi_muf = (const float*)d_in[5];
  const float* Wh_muf = (const float*)d_in[6];
  const float* bi_muf = (const float*)d_in[7];
  const float* bh_muf = (const float*)d_in[8];
  const float* Wi_mub = (const float*)d_in[9];
  const float* Wh_mub = (const float*)d_in[10];
  const float* bi_mub = (const float*)d_in[11];
  const float* bh_mub = (const float*)d_in[12];
  const float* Wi_sgf = (const float*)d_in[13];
  const float* Wh_sgf = (const float*)d_in[14];
  const float* bi_sgf = (const float*)d_in[15];
  const float* bh_sgf = (const float*)d_in[16];
  const float* Wi_sgb = (const float*)d_in[17];
  const float* Wh_sgb = (const float*)d_in[18];
  const float* bi_sgb = (const float*)d_in[19];
  const float* bh_sgb = (const float*)d_in[20];
  const float* Wi_d = (const float*)d_in[21];
  const float* Wh_d = (const float*)d_in[22];
  const float* bi_d = (const float*)d_in[23];
  const float* bh_d = (const float*)d_in[24];
  const float* Wl = (const float*)d_in[25];
  const float* bl = (const float*)d_in[26];

  float* out = (float*)d_out;
  float* ws = (float*)d_ws;
  float* out1 = ws;                 // 95232
  float* pool = ws + 95232;         // 23552
  float* henc = ws + 118784;        // 20480
  float* xwe  = ws + 139264;        // 1179648
  float* zbuf = ws + 1318912;       // 3072
  float* XW   = ws + 1321984;       // 884736  (total ~8.8 MB)

  k_conv1<<<372, 256, 0, stream>>>(input, W_c1, b_c1, out1);
  k_pool<<<92, 256, 0, stream>>>(out1, pool);
  k_conv2<<<80, 256, 0, stream>>>(pool, W_c2, b_c2, henc);
  k_xw_enc<<<4608, 256, 0, stream>>>(henc, Wi_muf, Wi_mub, Wi_sgf, Wi_sgb,
                                     bi_muf, bi_mub, bi_sgf, bi_sgb, xwe);
  k_gru_enc<<<4, 192, 0, stream>>>(Wh_muf, Wh_mub, Wh_sgf, Wh_sgb,
                                   bh_muf, bh_mub, bh_sgf, bh_sgb, xwe, out);
  k_z<<<12, 256, 0, stream>>>(out, zbuf);
  k_xw_dec<<<864, 128, 0, stream>>>(input, Wi_d, bi_d, XW);
  k_gru_dec<<<1, 384, 0, stream>>>(Wh_d, bh_d, XW, zbuf, Wl, bl, out);
}